// CRFLayer_48232482734977
// MI455X (gfx1250) — compile-verified
//
#include <hip/hip_runtime.h>
#include <hip/hip_bf16.h>
#include <math.h>

typedef __attribute__((ext_vector_type(2))) float v2f;
typedef __attribute__((ext_vector_type(8))) float v8f;

#define B_TOT  256
#define L_LEN  512
#define T_TAGS 128
#define BT     16     // batch rows per workgroup (WMMA M)
#define STR    132    // padded LDS row stride in dwords (132 % 64 == 4 -> conflict-free b64 A loads)

// ---------------------------------------------------------------------------
// Denominator: forward logsumexp scan, logsumexp mapped onto fp32 WMMA via
//   nxt[b,j] = em + M_b + log( sum_i exp(alpha[b,i]-M_b) * exp(trans[i,j]) )
// One workgroup per 16-batch tile; wave w owns dest-tag columns [16w,16w+16).
// exp(transitions) slice lives in 64 VGPRs per lane for the whole scan.
// K=128 reduction split into 4 independent WMMA accumulator chains for ILP
// (only 2 waves/SIMD are resident -> dependent-chain latency must be hidden
// inside the wave, not across waves).
// ---------------------------------------------------------------------------
__global__ __launch_bounds__(256)
void crf_forward_logz(const float* __restrict__ em,
                      const unsigned char* __restrict__ mask,
                      const float* __restrict__ startt,
                      const float* __restrict__ endt,
                      const float* __restrict__ trans,
                      float* __restrict__ logz)
{
    __shared__ __align__(16) float alpha[BT * STR];  // log-domain state
    __shared__ __align__(16) float P[BT * STR];      // exp(alpha - M) (WMMA A)
    __shared__ float Mv[BT];                         // per-row max

    const int tid  = threadIdx.x;
    const int lane = tid & 31;
    const int w    = tid >> 5;        // wave 0..7
    const int half = lane >> 4;       // 0 / 1 (K-split for A, M-split for D)
    const int lrow = lane & 15;
    const int b0   = blockIdx.x * BT;

    // B-matrix fragments: E[k, jcol] = exp(trans[k, jcol]), resident in VGPRs.
    // 4x16 B fragment layout: VGPR0 = rows {0,2} (by lane half), VGPR1 = rows {1,3}.
    const int jcol = w * 16 + lrow;
    float bE[64];
#pragma unroll
    for (int kc = 0; kc < 32; ++kc) {
        const int r0 = 4 * kc + 2 * half;
        bE[2 * kc + 0] = __expf(trans[(r0 + 0) * T_TAGS + jcol]);
        bE[2 * kc + 1] = __expf(trans[(r0 + 1) * T_TAGS + jcol]);
    }

    // alpha0 = start_transitions + emissions[:, 0, :]
    for (int i = tid; i < BT * T_TAGS; i += 256) {
        const int b = i >> 7, j = i & 127;
        alpha[b * STR + j] = startt[j] + em[((b0 + b) * L_LEN + 0) * T_TAGS + j];
    }
    __syncthreads();

    const int arow = tid >> 4;   // phase-A batch row (rows 2w, 2w+1 live in wave w)
    const int acol = tid & 15;

    for (int l = 1; l < L_LEN; ++l) {
        // ---- Phase A: per-row max and rescale into P ----
        float vals[8];
        float mx = -INFINITY;
#pragma unroll
        for (int k = 0; k < 8; ++k) {
            vals[k] = alpha[arow * STR + acol + 16 * k];
            mx = fmaxf(mx, vals[k]);
        }
#pragma unroll
        for (int off = 8; off >= 1; off >>= 1)
            mx = fmaxf(mx, __shfl_xor(mx, off, 32));   // stays within 16-lane half
#pragma unroll
        for (int k = 0; k < 8; ++k)
            P[arow * STR + acol + 16 * k] = __expf(vals[k] - mx);
        if (acol == 0) Mv[arow] = mx;
        if (l + 1 < L_LEN)
            __builtin_prefetch(&em[((b0 + arow) * L_LEN + l + 1) * T_TAGS + acol * 8], 0, 0);
        __syncthreads();

        // ---- Phase B: emissions in flight first, then 4 independent WMMA chains ----
        float ev[8];
#pragma unroll
        for (int r = 0; r < 8; ++r) {
            const int m = r + 8 * half;
            ev[r] = em[((b0 + m) * L_LEN + l) * T_TAGS + jcol];
        }

        v8f acc0 = {}, acc1 = {}, acc2 = {}, acc3 = {};
#pragma unroll
        for (int kc = 0; kc < 8; ++kc) {
            // A fragment: lanes 0-15 -> row=lrow, K={4k,4k+1}; lanes 16-31 -> K={4k+2,4k+3}
            const v2f a0 = *(const v2f*)&P[lrow * STR + 4 * (kc +  0) + 2 * half];
            const v2f a1 = *(const v2f*)&P[lrow * STR + 4 * (kc +  8) + 2 * half];
            const v2f a2 = *(const v2f*)&P[lrow * STR + 4 * (kc + 16) + 2 * half];
            const v2f a3 = *(const v2f*)&P[lrow * STR + 4 * (kc + 24) + 2 * half];
            v2f b0v; b0v.x = bE[2 * (kc +  0) + 0]; b0v.y = bE[2 * (kc +  0) + 1];
            v2f b1v; b1v.x = bE[2 * (kc +  8) + 0]; b1v.y = bE[2 * (kc +  8) + 1];
            v2f b2v; b2v.x = bE[2 * (kc + 16) + 0]; b2v.y = bE[2 * (kc + 16) + 1];
            v2f b3v; b3v.x = bE[2 * (kc + 24) + 0]; b3v.y = bE[2 * (kc + 24) + 1];
            acc0 = __builtin_amdgcn_wmma_f32_16x16x4_f32(false, a0, false, b0v, (short)0, acc0, false, false);
            acc1 = __builtin_amdgcn_wmma_f32_16x16x4_f32(false, a1, false, b1v, (short)0, acc1, false, false);
            acc2 = __builtin_amdgcn_wmma_f32_16x16x4_f32(false, a2, false, b2v, (short)0, acc2, false, false);
            acc3 = __builtin_amdgcn_wmma_f32_16x16x4_f32(false, a3, false, b3v, (short)0, acc3, false, false);
        }
        v8f acc = (acc0 + acc1) + (acc2 + acc3);

        // D layout: VGPR r -> M = r + 8*half, N = lrow
#pragma unroll
        for (int r = 0; r < 8; ++r) {
            const int   m   = r + 8 * half;
            const float nxt = ev[r] + Mv[m] + __logf(acc[r]);
            const float old = alpha[m * STR + jcol];
            const bool  mk  = mask[(b0 + m) * L_LEN + l] != 0;
            alpha[m * STR + jcol] = mk ? nxt : old;
        }
        __syncthreads();
    }

    // ---- log_z[b] = logsumexp_j(alpha[b,j] + end_transitions[j]) ----
    {
        float vals[8];
        float mx = -INFINITY;
#pragma unroll
        for (int k = 0; k < 8; ++k) {
            vals[k] = alpha[arow * STR + acol + 16 * k] + endt[acol + 16 * k];
            mx = fmaxf(mx, vals[k]);
        }
#pragma unroll
        for (int off = 8; off >= 1; off >>= 1)
            mx = fmaxf(mx, __shfl_xor(mx, off, 32));
        float s = 0.f;
#pragma unroll
        for (int k = 0; k < 8; ++k)
            s += __expf(vals[k] - mx);
#pragma unroll
        for (int off = 8; off >= 1; off >>= 1)
            s += __shfl_xor(s, off, 32);
        if (acol == 0) logz[b0 + arow] = mx + __logf(s);
    }
}

// ---------------------------------------------------------------------------
// Numerator: gold-path score, one block per batch, parallel over timesteps.
// ---------------------------------------------------------------------------
__global__ __launch_bounds__(256)
void crf_numerator(const float* __restrict__ em,
                   const int* __restrict__ tags,
                   const unsigned char* __restrict__ mask,
                   const float* __restrict__ startt,
                   const float* __restrict__ endt,
                   const float* __restrict__ trans,
                   float* __restrict__ num)
{
    const int b   = blockIdx.x;
    const int tid = threadIdx.x;
    __shared__ float reds[256];
    __shared__ int   redc[256];

    float s = 0.f;
    int cnt = 0;
    for (int l = tid; l < L_LEN; l += 256) {
        cnt += mask[b * L_LEN + l] ? 1 : 0;
        if (l >= 1) {
            const int   tp = tags[b * L_LEN + l - 1];
            const int   tc = tags[b * L_LEN + l];
            const float mv = mask[b * L_LEN + l] ? 1.f : 0.f;
            s += (trans[tp * T_TAGS + tc] + em[(b * L_LEN + l) * T_TAGS + tc]) * mv;
        }
    }
    reds[tid] = s;
    redc[tid] = cnt;
    __syncthreads();
    for (int off = 128; off; off >>= 1) {
        if (tid < off) {
            reds[tid] += reds[tid + off];
            redc[tid] += redc[tid + off];
        }
        __syncthreads();
    }
    if (tid == 0) {
        const int t0   = tags[b * L_LEN];
        const int last = tags[b * L_LEN + redc[0] - 1];   // seq_ends = sum(mask)-1
        num[b] = reds[0] + startt[t0] + em[(b * L_LEN) * T_TAGS + t0] + endt[last];
    }
}

// ---------------------------------------------------------------------------
// Final: -mean(llh) = mean(log_z - numerator)
// ---------------------------------------------------------------------------
__global__ __launch_bounds__(256)
void crf_finalize(const float* __restrict__ num,
                  const float* __restrict__ logz,
                  float* __restrict__ out)
{
    __shared__ float red[256];
    const int tid = threadIdx.x;
    red[tid] = logz[tid] - num[tid];
    __syncthreads();
    for (int off = 128; off; off >>= 1) {
        if (tid < off) red[tid] += red[tid + off];
        __syncthreads();
    }
    if (tid == 0) out[0] = red[0] / (float)B_TOT;
}

extern "C" void kernel_launch(void* const* d_in, const int* in_sizes, int n_in,
                              void* d_out, int out_size, void* d_ws, size_t ws_size,
                              hipStream_t stream)
{
    (void)in_sizes; (void)n_in; (void)out_size; (void)ws_size;
    const float*         em     = (const float*)d_in[0];
    const int*           tags   = (const int*)d_in[1];
    const unsigned char* mask   = (const unsigned char*)d_in[2];
    const float*         startt = (const float*)d_in[3];
    const float*         endt   = (const float*)d_in[4];
    const float*         trans  = (const float*)d_in[5];

    float* num  = (float*)d_ws;            // [256]
    float* logz = num + B_TOT;             // [256]

    crf_numerator   <<<B_TOT,      256, 0, stream>>>(em, tags, mask, startt, endt, trans, num);
    crf_forward_logz<<<B_TOT / BT, 256, 0, stream>>>(em, mask, startt, endt, trans, logz);
    crf_finalize    <<<1,          256, 0, stream>>>(num, logz, (float*)d_out);
}